// AdvancedPositionalBias_65498251264591
// MI455X (gfx1250) — compile-verified
//
#include <hip/hip_runtime.h>
#include <hip/hip_bf16.h>

typedef __attribute__((ext_vector_type(2))) float v2f;
typedef __attribute__((ext_vector_type(4))) float v4f;
typedef __attribute__((ext_vector_type(8))) float v8f;

#define B_DIM 16
#define S_DIM 256
#define DIN   1024
#define DOUT  1024

// One wave (32 lanes) computes one [16 x 16] output tile:
//   C[16(B) x 16(N)] = sum_K A[16x4] * Bmat[4x16]  via V_WMMA_F32_16X16X4_F32
// Block = 256 threads = 8 waves -> 8 adjacent N-tiles for one position s.
// Grid  = (64/8, S) = (8, 256).
__global__ __launch_bounds__(256, 1) void grouped_gemm_wmma_f32(
    const float* __restrict__ x,     // [B, S, DIN]
    const float* __restrict__ W,     // [S, DIN, DOUT]
    const float* __restrict__ bias,  // [S, DOUT]
    float* __restrict__ out)         // [B, S, DOUT]
{
    // Stage x[:, s, :] (16 x 1024 f32 = 64KB) in LDS, XOR-swizzled so that the
    // A-fragment ds_load_b64 (all lanes same column, different rows) is
    // bank-conflict free: col' = col ^ (4*m), rows stride 1024 (multiple of 64).
    __shared__ float ldsx[B_DIM * DIN];

    const int s = blockIdx.y;
    const int t = threadIdx.x;

    // Cooperative load: 16384 floats / 256 threads = 16 float4 per thread.
    #pragma unroll
    for (int r = 0; r < 16; ++r) {
        const int e = (r * 256 + t) * 4;     // flat element index, 16B aligned
        const int m = e >> 10;               // batch row 0..15
        const int i = e & 1023;              // din column, multiple of 4
        v4f v = *(const v4f*)(x + (((size_t)m * S_DIM + s) << 10) + i);
        const int col = i ^ ((m & 15) << 2); // XOR swizzle (keeps quads intact)
        *(v4f*)(ldsx + m * DIN + col) = v;
    }
    __syncthreads();

    const int lane = t & 31;
    const int wave = t >> 5;
    const int n0   = (blockIdx.x * 8 + wave) * 16;  // N-tile base in DOUT
    const int mn   = lane & 15;                     // A: row M; B: col N; C: col N
    const int kk   = (lane >> 4) * 2;               // K sub-pair select (0 or 2)

    // W walking pointer for this lane: element (k0+kk, n0+mn) of W[s].
    const float* wp = W + ((size_t)s * DIN) * DOUT + (size_t)kk * DOUT + n0 + mn;
    const float* arow = ldsx + mn * DIN;
    const int colbase = mn << 2;                    // XOR swizzle key for this row

    v8f c = {0.f, 0.f, 0.f, 0.f, 0.f, 0.f, 0.f, 0.f};

    #pragma unroll 4
    for (int k0 = 0; k0 < DIN; k0 += 4) {
        // A fragment: lane holds x[mn, s, k0+kk] and [.. k0+kk+1] (8B-aligned).
        const int col = (k0 ^ colbase) + kk;
        v2f a = *(const v2f*)(arow + col);

        // B fragment: lane holds W[s, k0+kk, n0+mn] and W[s, k0+kk+1, n0+mn].
        // W is single-use -> non-temporal so it doesn't evict x from L2.
        v2f b;
        b.x = __builtin_nontemporal_load(wp);
        b.y = __builtin_nontemporal_load(wp + DOUT);
        wp += 4 * DOUT;

        // D = A*B + C  (8 args: neg_a, A, neg_b, B, c_mod, C, reuse_a, reuse_b)
        c = __builtin_amdgcn_wmma_f32_16x16x4_f32(
                false, a, false, b, (short)0, c, false, false);
    }

    // Bias depends only on N (= mn for every C VGPR of this lane).
    const float bv = bias[s * DOUT + n0 + mn];

    // C layout: VGPR j -> M = j + 8*(lane>=16), N = lane&15.
    const int mbase = (lane >> 4) * 8;
    #pragma unroll
    for (int j = 0; j < 8; ++j) {
        const int m = mbase + j;
        out[(((size_t)m * S_DIM + s) << 10) + n0 + mn] = c[j] + bv;
    }
}

extern "C" void kernel_launch(void* const* d_in, const int* in_sizes, int n_in,
                              void* d_out, int out_size, void* d_ws, size_t ws_size,
                              hipStream_t stream) {
    const float* x    = (const float*)d_in[0];
    const float* W    = (const float*)d_in[1];
    const float* bias = (const float*)d_in[2];
    float* out        = (float*)d_out;

    dim3 grid(DOUT / 16 / 8, S_DIM);   // (8, 256)
    dim3 block(256);                   // 8 wave32 waves
    grouped_gemm_wmma_f32<<<grid, block, 0, stream>>>(x, W, bias, out);
}